// ImprovedBotnetDetectorV3Final_84834194030631
// MI455X (gfx1250) — compile-verified
//
#include <hip/hip_runtime.h>
#include <math.h>

// ---------------------------------------------------------------------------
// ImprovedBotnetDetectorV3Final for MI455X (gfx1250, wave32)
// GEMMs: fp32 WMMA (V_WMMA_F32_16X16X4_F32), branch-free inner loop with
// batched B-fragment loads (clause + staggered waits) and strength-reduced
// pointer arithmetic. Edge ops: L2-resident gathers + global f32 atomics.
// ---------------------------------------------------------------------------

typedef float v2f __attribute__((ext_vector_type(2)));
typedef float v8f __attribute__((ext_vector_type(8)));

#define N_NODES 50000
#define N_EDG   800000
#define E_TOT   (N_EDG + N_NODES)   // edges + self loops
#define ROWPAD  64                  // GEMM dest row padding (unguarded stores)

// ---------------- generic fp32 WMMA GEMM:  C[M,N] = A[M,K] @ B[K,N] (+bias)
// act: 0 = none, 1 = relu, 2 = tanh
// NOTE: C must have at least (round_up(M,64)) rows of backing store; rows
// >= M receive garbage (clamped-A products) and must never be read.
template <int NT>
__global__ __launch_bounds__(128) void k_gemm_wmma_f32(
    const float* __restrict__ A, const float* __restrict__ B,
    const float* __restrict__ bias, float* __restrict__ C,
    int M, int K, int N, int act) {
  const int lane = threadIdx.x & 31;
  const int wave = threadIdx.x >> 5;
  const int l15  = lane & 15;
  const int half = lane >> 4;          // 0: K pair {0,1}, 1: K pair {2,3}
  const int koff = half * 2;
  const int mbase = (blockIdx.y * 4 + wave) * 16;
  const int nbase = blockIdx.x * (16 * NT);
  int arow = mbase + l15;
  if (arow > M - 1) arow = M - 1;      // clamp: garbage rows stored to padding, never read

  const float* __restrict__ Ap = A + (size_t)arow * K + koff;   // += 4 per step
  const float* __restrict__ Bp = B + (size_t)koff * N + nbase + l15;  // += 4N

  const v8f vzero = {0.f, 0.f, 0.f, 0.f, 0.f, 0.f, 0.f, 0.f};
  v8f acc[NT];
#pragma unroll
  for (int t = 0; t < NT; ++t) acc[t] = vzero;

  for (int k0 = 0; k0 < K; k0 += 4) {
    const v2f a = *reinterpret_cast<const v2f*>(Ap);   // global_load_b64
    v2f b[NT];
#pragma unroll
    for (int t = 0; t < NT; ++t) {                     // one clause of 2*NT loads
      b[t].x = Bp[t * 16];
      b[t].y = Bp[(size_t)N + t * 16];
    }
#pragma unroll
    for (int t = 0; t < NT; ++t) {                     // staggered loadcnt waits
      acc[t] = __builtin_amdgcn_wmma_f32_16x16x4_f32(
          false, a, false, b[t], (short)0, acc[t], false, false);
    }
    Ap += 4;
    Bp += (size_t)4 * N;
  }
#pragma unroll
  for (int t = 0; t < NT; ++t) {
    const int col = nbase + t * 16 + l15;
    const float bv = bias ? bias[col] : 0.f;
#pragma unroll
    for (int r = 0; r < 8; ++r) {
      const int row = mbase + half * 8 + r;
      float v = acc[t][r] + bv;
      if (act == 1) v = fmaxf(v, 0.f);
      else if (act == 2) v = tanhf(v);
      C[(size_t)row * N + col] = v;     // unguarded: dest padded to 64-row tiles
    }
  }
}

// ---------------- small helpers ----------------
__global__ void k_fill(float* __restrict__ p, float v, int n) {
  int i = blockIdx.x * blockDim.x + threadIdx.x;
  if (i < n) p[i] = v;
}

// dmask = d-1 (d power of two)
__global__ void k_init_bias(float* __restrict__ out, const float* __restrict__ b,
                            int n, int dmask) {
  int i = blockIdx.x * blockDim.x + threadIdx.x;
  if (i < n) out[i] = b[i & dmask];
}

// x in place: bn (+ optional scalar prelu); dmask = d-1
__global__ void k_bn_prelu(float* __restrict__ x, const float* __restrict__ g,
                           const float* __restrict__ b, const float* __restrict__ mean,
                           const float* __restrict__ var, const float* aPtr,
                           int n, int dmask) {
  int i = blockIdx.x * blockDim.x + threadIdx.x;
  if (i >= n) return;
  int j = i & dmask;
  float v = (x[i] - mean[j]) * rsqrtf(var[j] + 1e-5f) * g[j] + b[j];
  if (aPtr) {
    float a = aPtr[0];
    v = (v >= 0.f) ? v : a * v;
  }
  x[i] = v;
}

__global__ void k_prelu(float* __restrict__ x, const float* aPtr, int n) {
  int i = blockIdx.x * blockDim.x + threadIdx.x;
  if (i >= n) return;
  float v = x[i];
  float a = aPtr[0];
  x[i] = (v >= 0.f) ? v : a * v;
}

// a_s[i] = sum_c h[i*C + c] * avec[(i & hmask)*C + c]; i in [0, n*H)
__global__ void k_headdot(const float* __restrict__ h, const float* __restrict__ avec,
                          float* __restrict__ out, int n, int hs, int C) {
  int i = blockIdx.x * blockDim.x + threadIdx.x;
  if (i >= n) return;
  int hh = i & ((1 << hs) - 1);
  const float* hp = h + (size_t)i * C;
  const float* ap = avec + hh * C;
  float s = 0.f;
  for (int c = 0; c < C; ++c) s += hp[c] * ap[c];
  out[i] = s;
}

__device__ __forceinline__ void atomicMaxF(float* addr, float val) {
  int* ia = reinterpret_cast<int*>(addr);
  int cur = __float_as_int(*addr);
  while (val > __int_as_float(cur)) {
    int prev = atomicCAS(ia, cur, __float_as_int(val));
    if (prev == cur) break;
    cur = prev;
  }
}

// per (edge, head): logit = leaky_relu(a_s[src]+a_d[dst]); segment max over dst
// H = 1<<hs
__global__ void k_edge_logit(const int* __restrict__ ei, const float* __restrict__ as,
                             const float* __restrict__ ad, float* __restrict__ elog,
                             float* __restrict__ segmax, int NE, int tot, int hs) {
  int i = blockIdx.x * blockDim.x + threadIdx.x;
  if (i >= tot) return;
  int e = i >> hs, hh = i & ((1 << hs) - 1);
  int s = (e < NE) ? ei[e] : (e - NE);
  int d = (e < NE) ? ei[NE + e] : (e - NE);
  float v = as[(s << hs) + hh] + ad[(d << hs) + hh];
  v = (v >= 0.f) ? v : 0.2f * v;
  elog[i] = v;
  atomicMaxF(segmax + (d << hs) + hh, v);
}

// per (edge, head): e = exp(logit - m[dst]); segment sum over dst
__global__ void k_edge_exp(const int* __restrict__ ei, float* __restrict__ elog,
                           const float* __restrict__ segmax, float* __restrict__ segsum,
                           int NE, int tot, int hs) {
  int i = blockIdx.x * blockDim.x + threadIdx.x;
  if (i >= tot) return;
  int e = i >> hs, hh = i & ((1 << hs) - 1);
  int d = (e < NE) ? ei[NE + e] : (e - NE);
  float v = expf(elog[i] - segmax[(d << hs) + hh]);
  elog[i] = v;
  atomicAdd(segsum + (d << hs) + hh, v);
}

// out[dst, h, c:c+4] += h[src, h, c:c+4] * alpha(edge, h)
// hc4s = log2(H*C/4), c4s = log2(C/4), hs = log2(H)
__global__ void k_gat_scatter(const int* __restrict__ ei, const float* __restrict__ eexp,
                              const float* __restrict__ segsum, const float* __restrict__ h,
                              float* __restrict__ out, int NE, int tot,
                              int hc4s, int c4s, int hs) {
  int i = blockIdx.x * blockDim.x + threadIdx.x;
  if (i >= tot) return;
  int e = i >> hc4s;
  int q = i & ((1 << hc4s) - 1);
  int hh = q >> c4s;
  int coff = (q & ((1 << c4s) - 1)) << 2;
  int s = (e < NE) ? ei[e] : (e - NE);
  int d = (e < NE) ? ei[NE + e] : (e - NE);
  float alpha = eexp[(e << hs) + hh] / (segsum[(d << hs) + hh] + 1e-16f);
  const int hc2 = hc4s + 2;  // log2(H*C)
  const float4 hv =
      *reinterpret_cast<const float4*>(h + ((size_t)s << hc2) + (hh << (c4s + 2)) + coff);
  float* op = out + ((size_t)d << hc2) + (hh << (c4s + 2)) + coff;
  atomicAdd(op + 0, hv.x * alpha);
  atomicAdd(op + 1, hv.y * alpha);
  atomicAdd(op + 2, hv.z * alpha);
  atomicAdd(op + 3, hv.w * alpha);
}

__global__ void k_degree(const int* __restrict__ ei, float* __restrict__ deg,
                         int NE, int tot) {
  int e = blockIdx.x * blockDim.x + threadIdx.x;
  if (e >= tot) return;
  int d = (e < NE) ? ei[NE + e] : (e - NE);
  atomicAdd(deg + d, 1.0f);
}

__global__ void k_rsqrt_deg(float* __restrict__ deg, int n) {
  int i = blockIdx.x * blockDim.x + threadIdx.x;
  if (i < n) deg[i] = rsqrtf(fmaxf(deg[i], 1.0f));
}

// out[dst, c:c+4] += h[src, c:c+4] * dis[src]*dis[dst]; c4s = log2(C/4)
__global__ void k_gcn_scatter(const int* __restrict__ ei, const float* __restrict__ h,
                              const float* __restrict__ dis, float* __restrict__ out,
                              int NE, int tot, int c4s) {
  int i = blockIdx.x * blockDim.x + threadIdx.x;
  if (i >= tot) return;
  int e = i >> c4s;
  int coff = (i & ((1 << c4s) - 1)) << 2;
  int s = (e < NE) ? ei[e] : (e - NE);
  int d = (e < NE) ? ei[NE + e] : (e - NE);
  float w = dis[s] * dis[d];
  const int cs = c4s + 2;  // log2(C)
  const float4 hv = *reinterpret_cast<const float4*>(h + ((size_t)s << cs) + coff);
  float* op = out + ((size_t)d << cs) + coff;
  atomicAdd(op + 0, hv.x * w);
  atomicAdd(op + 1, hv.y * w);
  atomicAdd(op + 2, hv.z * w);
  atomicAdd(op + 3, hv.w * w);
}

// wave-per-row LayerNorm over D=64 of (x + r)
__global__ void k_ln_residual64(const float* __restrict__ x, const float* __restrict__ r,
                                const float* __restrict__ g, const float* __restrict__ b,
                                float* __restrict__ out, int n) {
  int row = blockIdx.x * (blockDim.x >> 5) + (threadIdx.x >> 5);
  int lane = threadIdx.x & 31;
  if (row >= n) return;
  size_t base = (size_t)row * 64;
  float v0 = x[base + lane] + r[base + lane];
  float v1 = x[base + lane + 32] + r[base + lane + 32];
  float s = v0 + v1;
  float s2 = v0 * v0 + v1 * v1;
  for (int off = 16; off; off >>= 1) {
    s += __shfl_xor(s, off, 32);
    s2 += __shfl_xor(s2, off, 32);
  }
  float mu = s * (1.f / 64.f);
  float var = s2 * (1.f / 64.f) - mu * mu;
  float inv = rsqrtf(var + 1e-5f);
  out[base + lane]      = (v0 - mu) * inv * g[lane] + b[lane];
  out[base + lane + 32] = (v1 - mu) * inv * g[lane + 32] + b[lane + 32];
}

// out[n] = dot(X[n, :K], w) + b[0]   (wave per row, K <= 64)
__global__ void k_rowdot(const float* __restrict__ X, const float* __restrict__ w,
                         const float* __restrict__ bptr, float* __restrict__ out,
                         int n, int K) {
  int row = blockIdx.x * (blockDim.x >> 5) + (threadIdx.x >> 5);
  int lane = threadIdx.x & 31;
  if (row >= n) return;
  float s = 0.f;
  for (int k = lane; k < K; k += 32) s += X[(size_t)row * K + k] * w[k];
  for (int off = 16; off; off >>= 1) s += __shfl_xor(s, off, 32);
  if (lane == 0) out[row] = s + bptr[0];
}

// cross-modal softmax fusion: out[n,j] = sum_k softmax(sc)_k * h_k[n,j], D=128
__global__ void k_fuse(const float* __restrict__ h0, const float* __restrict__ h1,
                       const float* __restrict__ h2, const float* __restrict__ sc0,
                       const float* __restrict__ sc1, const float* __restrict__ sc2,
                       float* __restrict__ out, int n) {
  int i = blockIdx.x * blockDim.x + threadIdx.x;
  if (i >= n) return;
  int node = i >> 7;
  float s0 = sc0[node], s1 = sc1[node], s2 = sc2[node];
  float m = fmaxf(s0, fmaxf(s1, s2));
  float e0 = expf(s0 - m), e1 = expf(s1 - m), e2 = expf(s2 - m);
  float inv = 1.f / (e0 + e1 + e2);
  out[i] = (e0 * h0[i] + e1 * h1[i] + e2 * h2[i]) * inv;
}

// ---------------- host-side GEMM dispatch ----------------
static void gemm(const float* A, const float* B, const float* bias, float* C,
                 int M, int K, int N, int act, hipStream_t s) {
  dim3 blk(128);
  int gy = (M + 63) / 64;
  if (N % 64 == 0) {
    dim3 grd(N / 64, gy);
    k_gemm_wmma_f32<4><<<grd, blk, 0, s>>>(A, B, bias, C, M, K, N, act);
  } else {  // N % 32 == 0 (only N=32 here)
    dim3 grd(N / 32, gy);
    k_gemm_wmma_f32<2><<<grd, blk, 0, s>>>(A, B, bias, C, M, K, N, act);
  }
}

#define EL(tot) dim3((unsigned)(((tot) + 255) / 256)), dim3(256), 0, stream
#define WV(rows) dim3((unsigned)(((rows) + 3) / 4)), dim3(128), 0, stream

extern "C" void kernel_launch(void* const* d_in, const int* in_sizes, int n_in,
                              void* d_out, int out_size, void* d_ws, size_t ws_size,
                              hipStream_t stream) {
  (void)in_sizes; (void)n_in; (void)out_size; (void)ws_size;
  const int N = N_NODES, NE = N_EDG, ET = E_TOT;
  const int NP = N + ROWPAD;  // padded row count for GEMM destinations

  // ---- inputs (setup_inputs() dict insertion order, params recursively) ----
  const float* stat_x = (const float*)d_in[0];   // [N,64]
  const float* sem_x  = (const float*)d_in[1];   // [N,32]
  const float* str_x  = (const float*)d_in[2];   // [N,16]
  const int*   ei     = (const int*)d_in[3];     // [2,NE]
  #define IN(i) ((const float*)d_in[(i)])
  // gat1: 4..7  gat2: 8..11  bn_s1: 12..15  bn_s2: 16..19  prelu_s: 20
  // sem_in: 21,22  tl0: 23..34  tl1: 35..46  sem_out: 47,48
  // gcn1: 49,50  gcn2: 51,52  bn_g1: 53..56  bn_g2: 57..60  prelu_g: 61
  // cm: 62..67  cls: 68..75

  // ---- workspace layout (fp32, deterministic offsets) ----
  char* wsb = (char*)d_ws;
  size_t off = 0;
  auto alloc = [&](size_t nf) { float* p = (float*)(wsb + off); off += nf * sizeof(float); return p; };
  float* h1    = alloc((size_t)NP * 256);  // GEMM dest (GAT1/GAT2/GCN h), padded
  float* o1    = alloc((size_t)NP * 256);  // GAT1 out; reused: ta/tb/tc/go1
  float* as4   = alloc((size_t)N * 4);
  float* ad4   = alloc((size_t)N * 4);
  float* m4    = alloc((size_t)N * 4);
  float* s4    = alloc((size_t)N * 4);
  float* elog  = alloc((size_t)ET * 4);
  float* hstat = alloc((size_t)N * 128);   // scatter dest (exact N rows)
  float* hsem  = alloc((size_t)NP * 128);  // GEMM dest, padded
  float* hstr  = alloc((size_t)N * 128);   // scatter dest
  float* zbuf  = alloc((size_t)NP * 64);   // GEMM dest, padded
  float* t1    = alloc((size_t)NP * 128);  // GEMM dest, padded
  float* t2    = alloc((size_t)NP * 128);  // GEMM dest, padded
  float* deg   = alloc((size_t)N);
  float* gbuf  = alloc((size_t)NP * 64);   // GEMM dest, padded
  float* sc0   = alloc((size_t)N);
  float* sc1   = alloc((size_t)N);
  float* sc2   = alloc((size_t)N);
  // ta/tb/tc GEMM overflow rows land in the next sub-buffer but are always
  // rewritten/consumed in order; go1 is consumed before the fusion stage.
  float* ta = o1;                      // [N,64] tanh(hstat@Wa1)
  float* tb = o1 + (size_t)N * 64;
  float* tc = o1 + (size_t)N * 128;
  float* go1 = o1 + (size_t)N * 192;   // [N,64] gcn1 aggregation

  const float NEG_INF = -INFINITY;

  // ================= statistical encoder: GAT1 (4 heads x 64, concat) ======
  gemm(stat_x, IN(4), nullptr, h1, N, 64, 256, 0, stream);
  k_headdot<<<EL(N * 4)>>>(h1, IN(5), as4, N * 4, 2, 64);
  k_headdot<<<EL(N * 4)>>>(h1, IN(6), ad4, N * 4, 2, 64);
  k_fill<<<EL(N * 4)>>>(m4, NEG_INF, N * 4);
  k_fill<<<EL(N * 4)>>>(s4, 0.f, N * 4);
  k_init_bias<<<EL(N * 256)>>>(o1, IN(7), N * 256, 255);
  k_edge_logit<<<EL(ET * 4)>>>(ei, as4, ad4, elog, m4, NE, ET * 4, 2);
  k_edge_exp<<<EL(ET * 4)>>>(ei, elog, m4, s4, NE, ET * 4, 2);
  k_gat_scatter<<<EL(ET * 64)>>>(ei, elog, s4, h1, o1, NE, ET * 64, 6, 4, 2);
  k_bn_prelu<<<EL(N * 256)>>>(o1, IN(12), IN(13), IN(14), IN(15), IN(20), N * 256, 255);

  // ================= GAT2 (1 head x 128) ===================================
  gemm(o1, IN(8), nullptr, h1, N, 256, 128, 0, stream);
  k_headdot<<<EL(N)>>>(h1, IN(9), as4, N, 0, 128);
  k_headdot<<<EL(N)>>>(h1, IN(10), ad4, N, 0, 128);
  k_fill<<<EL(N)>>>(m4, NEG_INF, N);
  k_fill<<<EL(N)>>>(s4, 0.f, N);
  k_init_bias<<<EL(N * 128)>>>(hstat, IN(11), N * 128, 127);
  k_edge_logit<<<EL(ET)>>>(ei, as4, ad4, elog, m4, NE, ET, 0);
  k_edge_exp<<<EL(ET)>>>(ei, elog, m4, s4, NE, ET, 0);
  k_gat_scatter<<<EL(ET * 32)>>>(ei, elog, s4, h1, hstat, NE, ET * 32, 5, 5, 0);
  k_bn_prelu<<<EL(N * 128)>>>(hstat, IN(16), IN(17), IN(18), IN(19), nullptr, N * 128, 127);

  // ================= semantic encoder (transformer, seq_len=1) =============
  gemm(sem_x, IN(21), IN(22), zbuf, N, 32, 64, 0, stream);
  for (int L = 0; L < 2; ++L) {
    int b = 23 + L * 12;  // Wv bv Wo bo ln1g ln1b W1 b1 W2 b2 ln2g ln2b
    gemm(zbuf, IN(b + 0), IN(b + 1), t1, N, 64, 64, 0, stream);   // v = z@Wv+bv
    gemm(t1, IN(b + 2), IN(b + 3), t2, N, 64, 64, 0, stream);     // attn = v@Wo+bo
    k_ln_residual64<<<WV(N)>>>(zbuf, t2, IN(b + 4), IN(b + 5), zbuf, N);
    gemm(zbuf, IN(b + 6), IN(b + 7), t1, N, 64, 128, 1, stream);  // relu(z@W1+b1)
    gemm(t1, IN(b + 8), IN(b + 9), t2, N, 128, 64, 0, stream);    // ff@W2+b2
    k_ln_residual64<<<WV(N)>>>(zbuf, t2, IN(b + 10), IN(b + 11), zbuf, N);
  }
  gemm(zbuf, IN(47), IN(48), hsem, N, 64, 128, 0, stream);

  // ================= structural encoder (GCN) ==============================
  k_fill<<<EL(N)>>>(deg, 0.f, N);
  k_degree<<<EL(ET)>>>(ei, deg, NE, ET);
  k_rsqrt_deg<<<EL(N)>>>(deg, N);
  gemm(str_x, IN(49), nullptr, h1, N, 16, 64, 0, stream);
  k_init_bias<<<EL(N * 64)>>>(go1, IN(50), N * 64, 63);
  k_gcn_scatter<<<EL(ET * 16)>>>(ei, h1, deg, go1, NE, ET * 16, 4);
  k_bn_prelu<<<EL(N * 64)>>>(go1, IN(53), IN(54), IN(55), IN(56), IN(61), N * 64, 63);
  gemm(go1, IN(51), nullptr, h1, N, 64, 128, 0, stream);
  k_init_bias<<<EL(N * 128)>>>(hstr, IN(52), N * 128, 127);
  k_gcn_scatter<<<EL(ET * 32)>>>(ei, h1, deg, hstr, NE, ET * 32, 5);
  k_bn_prelu<<<EL(N * 128)>>>(hstr, IN(57), IN(58), IN(59), IN(60), nullptr, N * 128, 127);

  // ================= cross-modal attention fusion ==========================
  gemm(hstat, IN(62), IN(63), ta, N, 128, 64, 2, stream);  // tanh
  gemm(hsem,  IN(62), IN(63), tb, N, 128, 64, 2, stream);
  gemm(hstr,  IN(62), IN(63), tc, N, 128, 64, 2, stream);
  k_rowdot<<<WV(N)>>>(ta, IN(64), IN(65), sc0, N, 64);
  k_rowdot<<<WV(N)>>>(tb, IN(64), IN(65), sc1, N, 64);
  k_rowdot<<<WV(N)>>>(tc, IN(64), IN(65), sc2, N, 64);
  k_fuse<<<EL(N * 128)>>>(hstat, hsem, hstr, sc0, sc1, sc2, t1, N * 128);
  gemm(t1, IN(66), IN(67), t2, N, 128, 128, 0, stream);    // fused = ..@Wf+bf

  // ================= classifier ============================================
  gemm(t2, IN(68), IN(69), zbuf, N, 128, 64, 0, stream);
  k_prelu<<<EL(N * 64)>>>(zbuf, IN(70), N * 64);
  gemm(zbuf, IN(71), IN(72), gbuf, N, 64, 32, 0, stream);
  k_prelu<<<EL(N * 32)>>>(gbuf, IN(73), N * 32);
  k_rowdot<<<WV(N)>>>(gbuf, IN(74), IN(75), (float*)d_out, N, 32);
  #undef IN
}